// QCNN_Base_Shared_12756052869702
// MI455X (gfx1250) — compile-verified
//
#include <hip/hip_runtime.h>
#include <math.h>

// QCNN: 8 qubits, B=65536.
// Phase A (per-element, data-dependent): 4x [RY(x_q) on all q; CNOT ring] -> register sim, 8 lanes/element.
// Phase B (shared): CRX/U3 layers == fixed 256x256 unitary U -> real-expanded 512x512 f32 GEMM via WMMA.
// Epilogue: feats_q = sum sign_q |psi'|^2 for q in {3,7}; tiny MLP; sigmoid.

#define DIM 256
#define RD  512
#define NT  32          // batch elements per workgroup
#define STR 33          // padded LDS stride (in float2) to reduce bank conflicts

typedef __attribute__((ext_vector_type(2))) float v2f;
typedef __attribute__((ext_vector_type(8))) float v8f;

// ------------------------------------------------------------------
// Kernel 1: build W = real-expanded shared unitary (512x512), on device.
// Thread j evolves basis state |j> through the fixed CRX/U3 sequence.
// ------------------------------------------------------------------
__device__ __forceinline__ float2 cmulf(float2 a, float2 b) {
  return make_float2(a.x * b.x - a.y * b.y, a.x * b.y + a.y * b.x);
}

__global__ void build_w_kernel(const float* __restrict__ crx_theta,
                               const float* __restrict__ u3p,
                               float* __restrict__ W) {
  const int j = threadIdx.x;  // 0..255, single block
  float2 st[DIM];
  for (int a = 0; a < DIM; ++a) st[a] = make_float2(0.f, 0.f);
  st[j] = make_float2(1.f, 0.f);

  auto apply_crx = [&](int wc, int wt, float th) {
    float c = cosf(0.5f * th);
    float2 s = make_float2(0.f, -sinf(0.5f * th));  // -i sin
    int mc = 1 << (7 - wc), mt = 1 << (7 - wt);
    for (int a = 0; a < DIM; ++a) {
      if ((a & mc) && !(a & mt)) {
        int a1 = a | mt;
        float2 z0 = st[a], z1 = st[a1];
        float2 sz1 = cmulf(s, z1);
        float2 sz0 = cmulf(s, z0);
        st[a]  = make_float2(c * z0.x + sz1.x, c * z0.y + sz1.y);
        st[a1] = make_float2(sz0.x + c * z1.x, sz0.y + c * z1.y);
      }
    }
  };
  auto apply_u3 = [&](int w, float th, float ph, float lm) {
    float ct = cosf(0.5f * th), sv = sinf(0.5f * th);
    float2 el = make_float2(cosf(lm), sinf(lm));
    float2 ep = make_float2(cosf(ph), sinf(ph));
    float2 m00 = make_float2(ct, 0.f);
    float2 m01 = make_float2(-el.x * sv, -el.y * sv);
    float2 m10 = make_float2(ep.x * sv, ep.y * sv);
    float2 epl = cmulf(ep, el);
    float2 m11 = make_float2(epl.x * ct, epl.y * ct);
    int mt = 1 << (7 - w);
    for (int a = 0; a < DIM; ++a) {
      if (!(a & mt)) {
        int a1 = a | mt;
        float2 z0 = st[a], z1 = st[a1];
        float2 n0 = cmulf(m00, z0), t0 = cmulf(m01, z1);
        float2 n1 = cmulf(m10, z0), t1 = cmulf(m11, z1);
        st[a]  = make_float2(n0.x + t0.x, n0.y + t0.y);
        st[a1] = make_float2(n1.x + t1.x, n1.y + t1.y);
      }
    }
  };

  float th0 = crx_theta[0], th1 = crx_theta[1];
  // Layer 0: active = [0..7]
  apply_crx(0, 1, th0); apply_crx(2, 3, th0); apply_crx(4, 5, th0); apply_crx(6, 7, th0);
  apply_crx(1, 2, th0); apply_crx(3, 4, th0); apply_crx(5, 6, th0);
  apply_u3(1, u3p[0], u3p[1], u3p[2]);
  apply_u3(3, u3p[0], u3p[1], u3p[2]);
  apply_u3(5, u3p[0], u3p[1], u3p[2]);
  apply_u3(7, u3p[0], u3p[1], u3p[2]);
  // Layer 1: active = [1,3,5,7]
  apply_crx(1, 3, th1); apply_crx(5, 7, th1); apply_crx(3, 5, th1);
  apply_u3(3, u3p[3], u3p[4], u3p[5]);
  apply_u3(7, u3p[3], u3p[4], u3p[5]);

  // Real expansion: [Ur -Ui; Ui Ur], thread j owns columns j and 256+j.
  for (int a = 0; a < DIM; ++a) {
    float re = st[a].x, im = st[a].y;
    W[a * RD + j]               = re;
    W[a * RD + DIM + j]         = -im;
    W[(DIM + a) * RD + j]       = im;
    W[(DIM + a) * RD + DIM + j] = re;
  }
}

// ------------------------------------------------------------------
// Phase-A gate helpers: state = float sr[32], si[32] per lane.
// amplitude index a = (t << 5) | r, t = lane&7 (wires 0..2), r bits 4..0 (wires 3..7).
// ------------------------------------------------------------------
template <int P>
__device__ __forceinline__ void ry_reg(float (&sr)[32], float (&si)[32], float c, float s) {
#pragma unroll
  for (int r = 0; r < 32; ++r) {
    if (!(r & (1 << P))) {
      const int r1 = r | (1 << P);
      float a0r = sr[r], a1r = sr[r1];
      sr[r]  = c * a0r - s * a1r;
      sr[r1] = s * a0r + c * a1r;
      float a0i = si[r], a1i = si[r1];
      si[r]  = c * a0i - s * a1i;
      si[r1] = s * a0i + c * a1i;
    }
  }
}

template <int M>
__device__ __forceinline__ void ry_lane(float (&sr)[32], float (&si)[32], float c, float s, int lane) {
  float ss = (lane & M) ? s : -s;
#pragma unroll
  for (int r = 0; r < 32; ++r) {
    float pr = __shfl_xor(sr[r], M, 32);
    float pi = __shfl_xor(si[r], M, 32);
    sr[r] = fmaf(ss, pr, c * sr[r]);
    si[r] = fmaf(ss, pi, c * si[r]);
  }
}

// CNOT, control & target both lane bits
template <int MC, int MT>
__device__ __forceinline__ void cnot_ll(float (&sr)[32], float (&si)[32], int lane) {
  bool ctl = (lane & MC) != 0;
#pragma unroll
  for (int r = 0; r < 32; ++r) {
    float pr = __shfl_xor(sr[r], MT, 32);
    float pi = __shfl_xor(si[r], MT, 32);
    sr[r] = ctl ? pr : sr[r];
    si[r] = ctl ? pi : si[r];
  }
}

// CNOT q=2: control = lane bit0, target = register bit4
__device__ __forceinline__ void cnot_lr(float (&sr)[32], float (&si)[32], int lane) {
  bool f = (lane & 1) != 0;
#pragma unroll
  for (int r = 0; r < 16; ++r) {
    float a = sr[r], b = sr[r + 16];
    sr[r] = f ? b : a;  sr[r + 16] = f ? a : b;
    float ai = si[r], bi = si[r + 16];
    si[r] = f ? bi : ai; si[r + 16] = f ? ai : bi;
  }
}

// CNOT, both register bits: swap st[r] <-> st[r | 1<<PT] when control bit set
template <int PC, int PT>
__device__ __forceinline__ void cnot_rr(float (&sr)[32], float (&si)[32]) {
#pragma unroll
  for (int r = 0; r < 32; ++r) {
    if ((r & (1 << PC)) && !(r & (1 << PT))) {
      const int r1 = r | (1 << PT);
      float tr = sr[r]; sr[r] = sr[r1]; sr[r1] = tr;
      float ti = si[r]; si[r] = si[r1]; si[r1] = ti;
    }
  }
}

// CNOT q=7: control = register bit0 (odd r), target = lane bit2 (mask 4)
__device__ __forceinline__ void cnot_rl(float (&sr)[32], float (&si)[32]) {
#pragma unroll
  for (int r = 1; r < 32; r += 2) {
    sr[r] = __shfl_xor(sr[r], 4, 32);
    si[r] = __shfl_xor(si[r], 4, 32);
  }
}

// ------------------------------------------------------------------
// Kernel 2: fused sim + WMMA GEMM + readout + MLP. 256 threads, NT=32 elems/WG.
// ------------------------------------------------------------------
__global__ __launch_bounds__(256) void qcnn_kernel(
    const float* __restrict__ x, const float* __restrict__ W,
    const float* __restrict__ w1, const float* __restrict__ b1,
    const float* __restrict__ w2, const float* __restrict__ b2,
    float* __restrict__ out) {
  __shared__ float2 Sld[DIM * STR];   // state tile: [amp][elem], padded
  __shared__ float lfeat[NT][2];

  const int tid = threadIdx.x;
  const int lane = tid & 31;

  // ---------------- Phase A: register state-vector simulation ----------------
  {
    const int e = tid >> 3;      // local element 0..31
    const int t = tid & 7;       // lane-in-group (wires 0..2)
    const long b = (long)blockIdx.x * NT + e;

    float c8[8], s8[8];
#pragma unroll
    for (int w = 0; w < 8; ++w) {
      float a = 0.5f * x[b * 8 + w];
      c8[w] = cosf(a);
      s8[w] = sinf(a);
    }

    float sr[32], si[32];
#pragma unroll
    for (int r = 0; r < 32; ++r) { sr[r] = 0.f; si[r] = 0.f; }
    sr[0] = (t == 0) ? 1.f : 0.f;

    for (int cyc = 0; cyc < 4; ++cyc) {
      ry_lane<4>(sr, si, c8[0], s8[0], lane);  // wire 0
      ry_lane<2>(sr, si, c8[1], s8[1], lane);  // wire 1
      ry_lane<1>(sr, si, c8[2], s8[2], lane);  // wire 2
      ry_reg<4>(sr, si, c8[3], s8[3]);         // wire 3
      ry_reg<3>(sr, si, c8[4], s8[4]);         // wire 4
      ry_reg<2>(sr, si, c8[5], s8[5]);         // wire 5
      ry_reg<1>(sr, si, c8[6], s8[6]);         // wire 6
      ry_reg<0>(sr, si, c8[7], s8[7]);         // wire 7
      cnot_ll<4, 2>(sr, si, lane);  // (0,1)
      cnot_ll<2, 1>(sr, si, lane);  // (1,2)
      cnot_lr(sr, si, lane);        // (2,3)
      cnot_rr<4, 3>(sr, si);        // (3,4)
      cnot_rr<3, 2>(sr, si);        // (4,5)
      cnot_rr<2, 1>(sr, si);        // (5,6)
      cnot_rr<1, 0>(sr, si);        // (6,7)
      cnot_rl(sr, si);              // (7,0)
    }

#pragma unroll
    for (int r = 0; r < 32; ++r) {
      int amp = (t << 5) | r;
      Sld[amp * STR + e] = make_float2(sr[r], si[r]);
    }
  }
  if (tid < NT) { lfeat[tid][0] = 0.f; lfeat[tid][1] = 0.f; }
  __syncthreads();

  // ---------------- Phase B: 512x512 x 512xNT f32 WMMA GEMM ----------------
  const int wv = tid >> 5;           // 8 waves
  const int lm = lane & 15;
  const int lh = lane >> 4;          // 0/1: K half within fragment
  const int mbase = wv * 64;         // 64 output rows per wave (4 M-subtiles)
  const float* sp = reinterpret_cast<const float*>(Sld);
  const float2* W2 = reinterpret_cast<const float2*>(W);

  v8f acc[4][2];
  v8f z8 = {0.f, 0.f, 0.f, 0.f, 0.f, 0.f, 0.f, 0.f};
#pragma unroll
  for (int ms = 0; ms < 4; ++ms)
#pragma unroll
    for (int ns = 0; ns < 2; ++ns) acc[ms][ns] = z8;

  for (int k = 0; k < RD; k += 4) {
    const int kk0 = k + 2 * lh;
    v2f afr[4];
#pragma unroll
    for (int ms = 0; ms < 4; ++ms) {
      int row = mbase + ms * 16 + lm;
      float2 av = W2[(row * RD + kk0) >> 1];
      afr[ms].x = av.x; afr[ms].y = av.y;
    }
    v2f bfr[2];
#pragma unroll
    for (int ns = 0; ns < 2; ++ns) {
      int col = ns * 16 + lm;
      int amp0 = kk0 & 255;
      int amp1 = (kk0 + 1) & 255;
      int comp = (kk0 < DIM) ? 0 : 1;   // re rows then im rows (K chunk never straddles)
      bfr[ns].x = sp[(amp0 * STR + col) * 2 + comp];
      bfr[ns].y = sp[(amp1 * STR + col) * 2 + comp];
    }
#pragma unroll
    for (int ms = 0; ms < 4; ++ms)
#pragma unroll
      for (int ns = 0; ns < 2; ++ns)
        acc[ms][ns] = __builtin_amdgcn_wmma_f32_16x16x4_f32(
            false, afr[ms], false, bfr[ns], (short)0, acc[ms][ns], false, false);
  }

  // ---------------- Epilogue: feats = sum sign_q * v^2 ----------------
  float f3[2] = {0.f, 0.f}, f7[2] = {0.f, 0.f};
#pragma unroll
  for (int ms = 0; ms < 4; ++ms) {
#pragma unroll
    for (int j = 0; j < 8; ++j) {
      int row = mbase + ms * 16 + j + lh * 8;   // C layout: vgpr j -> M=j (lanes<16) / j+8
      int amp = row & 255;
      float s3 = (amp & 16) ? -1.f : 1.f;       // wire 3 <-> bit 4
      float s7 = (amp & 1) ? -1.f : 1.f;        // wire 7 <-> bit 0
#pragma unroll
      for (int ns = 0; ns < 2; ++ns) {
        float v = acc[ms][ns][j];
        float v2 = v * v;
        f3[ns] += s3 * v2;
        f7[ns] += s7 * v2;
      }
    }
  }
#pragma unroll
  for (int ns = 0; ns < 2; ++ns) {
    atomicAdd(&lfeat[ns * 16 + lm][0], f3[ns]);
    atomicAdd(&lfeat[ns * 16 + lm][1], f7[ns]);
  }
  __syncthreads();

  // ---------------- MLP + sigmoid ----------------
  if (tid < NT) {
    float f0 = lfeat[tid][0];
    float f1 = lfeat[tid][1];
    float o = b2[0];
#pragma unroll
    for (int j = 0; j < 10; ++j) {
      float h = tanhf(f0 * w1[j] + f1 * w1[10 + j] + b1[j]);
      o += h * w2[j];
    }
    long b = (long)blockIdx.x * NT + tid;
    out[b] = 1.f / (1.f + expf(-o));
  }
}

// ------------------------------------------------------------------
extern "C" void kernel_launch(void* const* d_in, const int* in_sizes, int n_in,
                              void* d_out, int out_size, void* d_ws, size_t ws_size,
                              hipStream_t stream) {
  const float* x   = (const float*)d_in[0];
  const float* crx = (const float*)d_in[1];
  const float* u3p = (const float*)d_in[2];
  const float* w1  = (const float*)d_in[3];
  const float* b1  = (const float*)d_in[4];
  const float* w2  = (const float*)d_in[5];
  const float* b2  = (const float*)d_in[6];
  float* out = (float*)d_out;
  float* W   = (float*)d_ws;               // 512*512 floats = 1 MB
  (void)n_in; (void)out_size; (void)ws_size;

  const int B = in_sizes[0] / 8;           // 65536

  build_w_kernel<<<1, 256, 0, stream>>>(crx, u3p, W);
  qcnn_kernel<<<B / NT, 256, 0, stream>>>(x, W, w1, b1, w2, b2, out);
}